// MambaVisionBlock_68212670595081
// MI455X (gfx1250) — compile-verified
//
#include <hip/hip_runtime.h>
#include <cstdint>
#include <cstddef>

// ---- problem constants (from reference) ------------------------------------
#define D_MODEL  192
#define D_STATE  16
#define D_CONV   4
#define D_INNER  384                    // EXPAND * D_MODEL
#define DT_RANK  12                     // (192+15)/16
#define NXDBC    (DT_RANK + 2*D_STATE)  // 44
#define BATCH    16
#define SEQLEN   4096                   // 64*64
#define MROWS    (BATCH * SEQLEN)       // 65536
#define NXZ      (2 * D_INNER)          // 768

typedef __attribute__((ext_vector_type(2))) float v2f;
typedef __attribute__((ext_vector_type(8))) float v8f;

__device__ __forceinline__ float sigmoidf_(float x) { return 1.0f / (1.0f + __expf(-x)); }
__device__ __forceinline__ float siluf_(float x)    { return x * sigmoidf_(x); }

// ---------------------------------------------------------------------------
// 1) LayerNorm over channels per pixel; emit sequence-major xn[m, 192]
//    x is (B, C, HW): channel reads are stride-HW but coalesced across threads.
// ---------------------------------------------------------------------------
__global__ void k_layernorm(const float* __restrict__ x,
                            const float* __restrict__ g,
                            const float* __restrict__ bb,
                            float* __restrict__ xn) {
  int p = blockIdx.x * blockDim.x + threadIdx.x;      // pixel = b*HW + l
  if (p >= MROWS) return;
  int b = p >> 12;
  int l = p & (SEQLEN - 1);
  const float* xp = x + (size_t)b * D_MODEL * SEQLEN + l;
  float s = 0.f, ss = 0.f;
  for (int c = 0; c < D_MODEL; ++c) {
    float v = xp[(size_t)c * SEQLEN];
    s += v; ss += v * v;
  }
  float mu  = s * (1.0f / D_MODEL);
  float var = ss * (1.0f / D_MODEL) - mu * mu;
  float rs  = rsqrtf(var + 1e-5f);
  float* op = xn + (size_t)p * D_MODEL;
  for (int c = 0; c < D_MODEL; ++c) {
    float v = xp[(size_t)c * SEQLEN];
    op[c] = (v - mu) * rs * g[c] + bb[c];
  }
}

// ---------------------------------------------------------------------------
// 2) fp32-exact WMMA GEMM: C[M,N] = A[M,K] * B[K,N]   (row-major)
//    One wave computes a 16(M) x 64(N) tile; K stepped by 4 via
//    v_wmma_f32_16x16x4_f32. M%16==0, N%64==0, K%4==0.
//    A frag (ISA 7.12.2): lane<16 -> row=lane, holds K=k,k+1;
//                         lane>=16 -> row=lane-16, holds K=k+2,k+3.
//    B frag mirrored (lane = column, VGPRs hold K rows).
//    C/D: vgpr r -> row r + 8*(lane>>4), col = lane&15.
// ---------------------------------------------------------------------------
__global__ void k_gemm_f32_wmma(const float* __restrict__ A,
                                const float* __restrict__ Bw,
                                float* __restrict__ C,
                                int M, int N, int K) {
  const int lane = threadIdx.x;            // 0..31
  const int m0   = (blockIdx.x * blockDim.y + threadIdx.y) * 16;
  const int n0   = blockIdx.y * 64;
  if (m0 >= M) return;                     // uniform per wave
  const int half = lane >> 4;
  const int lm   = lane & 15;

  const float* Arow = A + (size_t)(m0 + lm) * K + 2 * half;
  v8f acc0 = {}, acc1 = {}, acc2 = {}, acc3 = {};

  for (int k = 0; k < K; k += 4) {
    v2f a; a.x = Arow[k]; a.y = Arow[k + 1];
    const float* Bp = Bw + (size_t)(k + 2 * half) * N + n0 + lm;
    v2f b0, b1, b2, b3;
    b0.x = Bp[0];  b0.y = Bp[N];
    b1.x = Bp[16]; b1.y = Bp[N + 16];
    b2.x = Bp[32]; b2.y = Bp[N + 32];
    b3.x = Bp[48]; b3.y = Bp[N + 48];
    acc0 = __builtin_amdgcn_wmma_f32_16x16x4_f32(false, a, false, b0, (short)0, acc0, false, false);
    acc1 = __builtin_amdgcn_wmma_f32_16x16x4_f32(false, a, false, b1, (short)0, acc1, false, false);
    acc2 = __builtin_amdgcn_wmma_f32_16x16x4_f32(false, a, false, b2, (short)0, acc2, false, false);
    acc3 = __builtin_amdgcn_wmma_f32_16x16x4_f32(false, a, false, b3, (short)0, acc3, false, false);
  }

  float* Crow = C + (size_t)m0 * N + n0;
  #pragma unroll
  for (int r = 0; r < 8; ++r) {
    size_t off = (size_t)(r + 8 * half) * N + lm;
    Crow[off]      = acc0[r];
    Crow[off + 16] = acc1[r];
    Crow[off + 32] = acc2[r];
    Crow[off + 48] = acc3[r];
  }
}

// ---------------------------------------------------------------------------
// 7) same WMMA GEMM, fused residual add + transposed NCHW store:
//    out[b, c, l] = (yg @ W_out)[m=b*L+l, n=c] + x[b, c, l]
// ---------------------------------------------------------------------------
__global__ void k_gemm_out_wmma(const float* __restrict__ A,   // yg [M, 384]
                                const float* __restrict__ Bw,  // W_out [384,192]
                                const float* __restrict__ xres,// x (B,C,HW)
                                float* __restrict__ out) {
  const int lane = threadIdx.x;
  const int m0   = (blockIdx.x * blockDim.y + threadIdx.y) * 16;
  const int n0   = blockIdx.y * 64;
  const int M = MROWS, N = D_MODEL, K = D_INNER;
  if (m0 >= M) return;
  const int half = lane >> 4;
  const int lm   = lane & 15;

  const float* Arow = A + (size_t)(m0 + lm) * K + 2 * half;
  v8f acc0 = {}, acc1 = {}, acc2 = {}, acc3 = {};

  for (int k = 0; k < K; k += 4) {
    v2f a; a.x = Arow[k]; a.y = Arow[k + 1];
    const float* Bp = Bw + (size_t)(k + 2 * half) * N + n0 + lm;
    v2f b0, b1, b2, b3;
    b0.x = Bp[0];  b0.y = Bp[N];
    b1.x = Bp[16]; b1.y = Bp[N + 16];
    b2.x = Bp[32]; b2.y = Bp[N + 32];
    b3.x = Bp[48]; b3.y = Bp[N + 48];
    acc0 = __builtin_amdgcn_wmma_f32_16x16x4_f32(false, a, false, b0, (short)0, acc0, false, false);
    acc1 = __builtin_amdgcn_wmma_f32_16x16x4_f32(false, a, false, b1, (short)0, acc1, false, false);
    acc2 = __builtin_amdgcn_wmma_f32_16x16x4_f32(false, a, false, b2, (short)0, acc2, false, false);
    acc3 = __builtin_amdgcn_wmma_f32_16x16x4_f32(false, a, false, b3, (short)0, acc3, false, false);
  }

  #pragma unroll
  for (int r = 0; r < 8; ++r) {
    int m = m0 + r + 8 * half;
    int b = m >> 12;
    int l = m & (SEQLEN - 1);
    size_t rowbase = ((size_t)b * D_MODEL) * SEQLEN + l;   // + c*SEQLEN
    #pragma unroll
    for (int nn = 0; nn < 4; ++nn) {
      int c = n0 + nn * 16 + lm;
      size_t idx = rowbase + (size_t)c * SEQLEN;
      float acc = (nn == 0) ? acc0[r] : (nn == 1) ? acc1[r] : (nn == 2) ? acc2[r] : acc3[r];
      out[idx] = acc + xres[idx];
    }
  }
}

// ---------------------------------------------------------------------------
// 3) depthwise causal conv (k=4) + bias + SiLU:  u[m, d]
//    xi = xz[:, 0:384]; consecutive threads = consecutive d (coalesced).
// ---------------------------------------------------------------------------
__global__ void k_conv_silu(const float* __restrict__ xz,
                            const float* __restrict__ cw,
                            const float* __restrict__ cb,
                            float* __restrict__ u) {
  int i = blockIdx.x * blockDim.x + threadIdx.x;       // over MROWS*D_INNER
  if (i >= MROWS * D_INNER) return;
  int d = i % D_INNER;
  int m = i / D_INNER;
  int l = m & (SEQLEN - 1);
  const float* w = cw + d * D_CONV;
  float acc = cb[d];
  #pragma unroll
  for (int j = 0; j < D_CONV; ++j) {
    int lj = l - (D_CONV - 1) + j;                      // causal left pad 3
    if (lj >= 0) acc += w[j] * xz[(size_t)(m - (D_CONV - 1) + j) * NXZ + d];
  }
  u[(size_t)m * D_INNER + d] = siluf_(acc);
}

// ---------------------------------------------------------------------------
// 4) xdbc = u @ W_x   (N=44: too ragged for 16-wide WMMA; memory-bound VALU)
// ---------------------------------------------------------------------------
__global__ void k_xdbc(const float* __restrict__ u,
                       const float* __restrict__ Wx,
                       float* __restrict__ xdbc) {
  int i = blockIdx.x * blockDim.x + threadIdx.x;       // over MROWS*NXDBC
  if (i >= MROWS * NXDBC) return;
  int n = i % NXDBC;
  int m = i / NXDBC;
  const float* ur = u + (size_t)m * D_INNER;
  float acc = 0.f;
  for (int k = 0; k < D_INNER; ++k) acc += ur[k] * Wx[k * NXDBC + n];
  xdbc[i] = acc;
}

// ---------------------------------------------------------------------------
// 5) dt = softplus(dt_r @ W_dt + b_dt)   (K=12, memory-bound)
// ---------------------------------------------------------------------------
__global__ void k_dt(const float* __restrict__ xdbc,
                     const float* __restrict__ Wdt,
                     const float* __restrict__ bdt,
                     float* __restrict__ dt) {
  int i = blockIdx.x * blockDim.x + threadIdx.x;       // over MROWS*D_INNER
  if (i >= MROWS * D_INNER) return;
  int d = i % D_INNER;
  int m = i / D_INNER;
  const float* xr = xdbc + (size_t)m * NXDBC;
  float acc = bdt[d];
  #pragma unroll
  for (int r = 0; r < DT_RANK; ++r) acc += xr[r] * Wdt[r * D_INNER + d];
  dt[i] = (acc > 20.f) ? acc : log1pf(__expf(acc));
}

// ---------------------------------------------------------------------------
// 6) selective scan: one thread per (b, d), 16 states in registers.
//    Fused: y = scan + u*Dp, gated by silu(z); writes yg (aliased onto u:
//    each (m,d) element is read into a register before being overwritten,
//    by the same thread — safe).
// ---------------------------------------------------------------------------
__global__ void k_scan(const float* __restrict__ u,
                       const float* __restrict__ dt,
                       const float* __restrict__ xdbc,
                       const float* __restrict__ xz,
                       const float* __restrict__ A_log,
                       const float* __restrict__ Dp,
                       float* __restrict__ yg) {
  int t = blockIdx.x * blockDim.x + threadIdx.x;
  if (t >= BATCH * D_INNER) return;
  int d = t % D_INNER;
  int b = t / D_INNER;

  float Arow[D_STATE];
  #pragma unroll
  for (int s = 0; s < D_STATE; ++s) Arow[s] = -__expf(A_log[d * D_STATE + s]);
  float Dd = Dp[d];
  float h[D_STATE];
  #pragma unroll
  for (int s = 0; s < D_STATE; ++s) h[s] = 0.f;

  size_t base = (size_t)b * SEQLEN;
  for (int l = 0; l < SEQLEN; ++l) {
    size_t m = base + l;
    float ut  = u[m * D_INNER + d];
    float dtt = dt[m * D_INNER + d];
    float zt  = xz[m * NXZ + D_INNER + d];
    const float* bc = xdbc + m * NXDBC + DT_RANK;      // B_t[0..15], C_t[16..31]
    float dbu = dtt * ut;
    float y = 0.f;
    #pragma unroll
    for (int s = 0; s < D_STATE; ++s) {
      float dA = __expf(dtt * Arow[s]);
      h[s] = h[s] * dA + dbu * bc[s];
      y += h[s] * bc[D_STATE + s];
    }
    y += ut * Dd;
    y *= siluf_(zt);
    yg[m * D_INNER + d] = y;
  }
}

// ---------------------------------------------------------------------------
extern "C" void kernel_launch(void* const* d_in, const int* in_sizes, int n_in,
                              void* d_out, int out_size, void* d_ws, size_t ws_size,
                              hipStream_t stream) {
  const float* x      = (const float*)d_in[0];
  const float* ln_g   = (const float*)d_in[1];
  const float* ln_b   = (const float*)d_in[2];
  const float* W_in   = (const float*)d_in[3];
  const float* conv_w = (const float*)d_in[4];
  const float* conv_b = (const float*)d_in[5];
  const float* W_x    = (const float*)d_in[6];
  const float* W_dt   = (const float*)d_in[7];
  const float* b_dt   = (const float*)d_in[8];
  const float* A_log  = (const float*)d_in[9];
  const float* Dp     = (const float*)d_in[10];
  const float* W_out  = (const float*)d_in[11];
  float* out = (float*)d_out;

  // workspace layout (floats):
  //   xn  [MROWS*192]  -> reused for xdbc [MROWS*44] after gemm_in consumes it
  //   xz  [MROWS*768]
  //   u   [MROWS*384]  -> reused as yg by the scan (read-before-write per elem)
  //   dt  [MROWS*384]
  float* xn  = (float*)d_ws;
  float* xz  = xn + (size_t)MROWS * D_MODEL;
  float* u   = xz + (size_t)MROWS * NXZ;
  float* dtb = u  + (size_t)MROWS * D_INNER;
  float* xd  = xn;                                   // alias (xn dead after gemm_in)
  float* yg  = u;                                    // alias (safe, see k_scan)

  (void)in_sizes; (void)n_in; (void)out_size; (void)ws_size;

  // 1) layernorm -> xn [M, 192]
  k_layernorm<<<MROWS / 256, 256, 0, stream>>>(x, ln_g, ln_b, xn);

  // 2) xz = xn @ W_in   (M=65536, K=192, N=768), fp32 WMMA
  {
    dim3 grid(MROWS / 64, NXZ / 64);
    dim3 blk(32, 4);
    k_gemm_f32_wmma<<<grid, blk, 0, stream>>>(xn, W_in, xz, MROWS, NXZ, D_MODEL);
  }

  // 3) u = silu(causal_conv(xi) + b)
  k_conv_silu<<<(MROWS * D_INNER) / 256, 256, 0, stream>>>(xz, conv_w, conv_b, u);

  // 4) xdbc = u @ W_x   [M, 44]
  k_xdbc<<<(MROWS * NXDBC + 255) / 256, 256, 0, stream>>>(u, W_x, xd);

  // 5) dt = softplus(dt_r @ W_dt + b_dt)  [M, 384]
  k_dt<<<(MROWS * D_INNER) / 256, 256, 0, stream>>>(xd, W_dt, b_dt, dtb);

  // 6) selective scan + Dp skip + silu(z) gate -> yg [M, 384]
  k_scan<<<(BATCH * D_INNER) / 128, 128, 0, stream>>>(u, dtb, xd, xz, A_log, Dp, yg);

  // 7) out = yg @ W_out + residual, stored NCHW  (M=65536, K=384, N=192), fp32 WMMA
  {
    dim3 grid(MROWS / 64, D_MODEL / 64);
    dim3 blk(32, 4);
    k_gemm_out_wmma<<<grid, blk, 0, stream>>>(yg, W_out, x, out);
  }
}